// GraphImportanceGNN_54657753809364
// MI455X (gfx1250) — compile-verified
//
#include <hip/hip_runtime.h>

#define NN 100000
#define NE 1600000
#define CH 128            // IN_CH == HID == 128
#define MTILES (NN / 16)  // 6250, exact

typedef float v2f __attribute__((ext_vector_type(2)));
typedef float v8f __attribute__((ext_vector_type(8)));

// Native CDNA5 f32 atomic-add, no return (tracked with STOREcnt), device scope
// so the RMW resolves at L2 (shared across all WGPs).
__device__ __forceinline__ void atom_add_f32(float* p, float v) {
  asm volatile("global_atomic_add_f32 %0, %1, off scope:SCOPE_DEV"
               :: "v"(p), "v"(v) : "memory");
}

// ---------------- degree / normalization ----------------
__global__ void k_deg_init(float* deg) {
  int i = blockIdx.x * blockDim.x + threadIdx.x;
  if (i < NN) deg[i] = 1.0f;  // self-loop
}

__global__ void k_deg_accum(const int* __restrict__ dst, float* deg) {
  int e = blockIdx.x * blockDim.x + threadIdx.x;
  if (e < NE) atom_add_f32(deg + dst[e], 1.0f);
}

__global__ void k_rsqrt(float* degdis) {
  int i = blockIdx.x * blockDim.x + threadIdx.x;
  if (i < NN) degdis[i] = rsqrtf(degdis[i]);
}

// ---------------- fp32 GEMM via V_WMMA_F32_16X16X4_F32 ----------------
// C[M x 128] = op(A)[M x 128] * W[128 x 128], row-major fp32.
// op(A) = relu(A + bias) when FUSE_RELU_BIAS (GCN layer-2 input), else A.
// One wave32 computes a 16-row x 128-col strip (8 N-tiles, K-loop of 32x4).
template <bool FUSE_RELU_BIAS>
__global__ __launch_bounds__(256) void k_gemm_wmma(const float* __restrict__ A,
                                                   const float* __restrict__ W,
                                                   const float* __restrict__ bias,
                                                   float* __restrict__ C) {
  const int lane = threadIdx.x & 31;
  const int wave = threadIdx.x >> 5;
  const int rowTile = blockIdx.x * 8 + wave;
  if (rowTile >= MTILES) return;  // wave-uniform: EXEC all-1s inside WMMA

  const int m0 = rowTile * 16;
  const int mrow = m0 + (lane & 15);     // A: lane -> M row
  const int kl = (lane >> 4) * 2;        // K sub-pair: 0 (lanes 0-15) or 2
  const int ncol = lane & 15;            // B: column within 16-wide tile

  v8f acc[8] = {};

  const float* __restrict__ arow = A + (size_t)mrow * CH + kl;
  for (int k0 = 0; k0 < CH; k0 += 4) {
    v2f a;
    a.x = arow[k0];
    a.y = arow[k0 + 1];
    if (FUSE_RELU_BIAS) {
      float2 bb = *(const float2*)(bias + k0 + kl);  // 8B-aligned, L0-resident
      a.x = fmaxf(a.x + bb.x, 0.0f);
      a.y = fmaxf(a.y + bb.y, 0.0f);
    }
    const float* __restrict__ wrow = W + (size_t)(k0 + kl) * CH + ncol;
#pragma unroll
    for (int t = 0; t < 8; ++t) {
      v2f b;
      b.x = wrow[t * 16];        // B[k0+kl    ][n0+ncol]
      b.y = wrow[CH + t * 16];   // B[k0+kl + 1][n0+ncol]
      acc[t] = __builtin_amdgcn_wmma_f32_16x16x4_f32(
          false, a, false, b, (short)0, acc[t], false, false);
    }
  }

  // C/D layout: VGPR r -> row m0 + r (lanes 0-15) or m0 + 8 + r (lanes 16-31)
  const int orow = m0 + (lane >> 4) * 8;
  float* cbase = C + (size_t)orow * CH + (lane & 15);
#pragma unroll
  for (int t = 0; t < 8; ++t) {
    float* cp = cbase + t * 16;
#pragma unroll
    for (int r = 0; r < 8; ++r) cp[(size_t)r * CH] = acc[t][r];
  }
}

// ---------------- self-loop init: agg = h * dis^2 ----------------
__global__ void k_self_init(const float* __restrict__ h,
                            const float* __restrict__ dis,
                            float* __restrict__ agg) {
  int idx = blockIdx.x * blockDim.x + threadIdx.x;  // NN*32 threads
  int node = idx >> 5;
  int c4 = idx & 31;
  if (node >= NN) return;
  float w = dis[node];
  w = w * w;
  float4 v = ((const float4*)(h + (size_t)node * CH))[c4];
  v.x *= w; v.y *= w; v.z *= w; v.w *= w;
  ((float4*)(agg + (size_t)node * CH))[c4] = v;
}

// ---------------- edge scatter: agg[dst] += h[src] * dis[src]*dis[dst] ----
// One wave32 per edge: lane c handles channels [4c, 4c+3] (coalesced 512B).
// Edge index is wave-uniform -> force it scalar so src/dst/dis ride SMEM.
__global__ __launch_bounds__(256) void k_edge_scatter(
    const float* __restrict__ h, const int* __restrict__ src,
    const int* __restrict__ dst, const float* __restrict__ dis,
    float* __restrict__ agg) {
  int e = __builtin_amdgcn_readfirstlane(blockIdx.x * 8 + (threadIdx.x >> 5));
  if (e >= NE) return;
  int lane = threadIdx.x & 31;
  int s = src[e];
  int d = dst[e];
  float nm = dis[s] * dis[d];
  float4 v = ((const float4*)(h + (size_t)s * CH))[lane];
  float* ap = agg + (size_t)d * CH + lane * 4;
  atom_add_f32(ap + 0, v.x * nm);
  atom_add_f32(ap + 1, v.y * nm);
  atom_add_f32(ap + 2, v.z * nm);
  atom_add_f32(ap + 3, v.w * nm);
}

// ------- fused epilogue: out[i] = relu(agg2[i]+b2) . Wout + bout ---------
// One wave32 per node: lane handles 4 channels, shuffle-reduce across wave.
__global__ __launch_bounds__(256) void k_bias_relu_dot(
    const float* __restrict__ agg, const float* __restrict__ b,
    const float* __restrict__ wout, const float* __restrict__ bout,
    float* __restrict__ out) {
  int node = blockIdx.x * 8 + (threadIdx.x >> 5);
  if (node >= NN) return;
  int lane = threadIdx.x & 31;
  float4 v = ((const float4*)(agg + (size_t)node * CH))[lane];
  float4 bb = ((const float4*)b)[lane];
  float4 w = ((const float4*)wout)[lane];
  float s = fmaxf(v.x + bb.x, 0.0f) * w.x + fmaxf(v.y + bb.y, 0.0f) * w.y +
            fmaxf(v.z + bb.z, 0.0f) * w.z + fmaxf(v.w + bb.w, 0.0f) * w.w;
#pragma unroll
  for (int off = 16; off > 0; off >>= 1) s += __shfl_xor(s, off, 32);
  if (lane == 0) out[node] = s + bout[0];
}

extern "C" void kernel_launch(void* const* d_in, const int* in_sizes, int n_in,
                              void* d_out, int out_size, void* d_ws, size_t ws_size,
                              hipStream_t stream) {
  const float* x    = (const float*)d_in[0];
  const int*   ei   = (const int*)d_in[1];   // [2, NE]: row0=src, row1=dst
  const float* W1   = (const float*)d_in[2];
  const float* b1   = (const float*)d_in[3];
  const float* W2   = (const float*)d_in[4];
  const float* b2   = (const float*)d_in[5];
  const float* Wout = (const float*)d_in[6];
  const float* bout = (const float*)d_in[7];
  float* out = (float*)d_out;

  const int* src = ei;
  const int* dst = ei + NE;

  // workspace: dis (512KB slot) | bufH (51.2MB) | bufA (51.2MB)
  char* ws = (char*)d_ws;
  float* dis  = (float*)ws;
  float* bufH = (float*)(ws + (512u << 10));
  float* bufA = (float*)(ws + (512u << 10) + (size_t)NN * CH * sizeof(float));

  const int B = 256;
  dim3 blkNode((NN + B - 1) / B);
  dim3 blkEdge((NE + B - 1) / B);
  dim3 blkNC((NN * 32 + B - 1) / B);       // node*channel4 elementwise
  dim3 blkEdgeWave((NE + 7) / 8);          // wave per edge
  dim3 blkNodeWave((NN + 7) / 8);          // wave per node
  dim3 blkGemm((MTILES + 7) / 8);          // wave per 16-row strip

  // normalization: dis = rsqrt(1 + indegree)
  k_deg_init<<<blkNode, B, 0, stream>>>(dis);
  k_deg_accum<<<blkEdge, B, 0, stream>>>(dst, dis);
  k_rsqrt<<<blkNode, B, 0, stream>>>(dis);

  // layer 1: H1 = X @ W1 ; agg1 = normalized (A+I) aggregation of H1
  k_gemm_wmma<false><<<blkGemm, B, 0, stream>>>(x, W1, nullptr, bufH);
  k_self_init<<<blkNC, B, 0, stream>>>(bufH, dis, bufA);
  k_edge_scatter<<<blkEdgeWave, B, 0, stream>>>(bufH, src, dst, dis, bufA);

  // layer 2: H2 = relu(agg1 + b1) @ W2 (relu+bias fused into GEMM A-load)
  k_gemm_wmma<true><<<blkGemm, B, 0, stream>>>(bufA, W2, b1, bufH);
  k_self_init<<<blkNC, B, 0, stream>>>(bufH, dis, bufA);
  k_edge_scatter<<<blkEdgeWave, B, 0, stream>>>(bufH, src, dst, dis, bufA);

  // fused bias + relu + readout (no intermediate store of relu'd features)
  k_bias_relu_dot<<<blkNodeWave, B, 0, stream>>>(bufA, b2, Wout, bout, out);
}